// YOLOWrapper_472446403219
// MI455X (gfx1250) — compile-verified
//
#include <hip/hip_runtime.h>
#include <hip/hip_bf16.h>
#include <stdint.h>

#define BATCH 32
#define NPRED 25200
#define NCLS  20
#define ROW   25
#define KSEL  1024
#define MAXDET 300
#define CONF_T 0.001f
#define IOU_T  0.45f
#define MAXWH  4096.0f
#define NBINS  16384
#define CAP    2048
#define TILE_ROWS 256
#define NSLAB 8
#define SLAB_ROWS 128

typedef __attribute__((ext_vector_type(2))) float v2f;
typedef __attribute__((ext_vector_type(8))) float v8f;
typedef __attribute__((ext_vector_type(4))) int   v4i;

#if __has_builtin(__builtin_amdgcn_global_load_async_to_lds_b128)
#define HAVE_ASYNC_LDS 1
#endif

// The async-to-LDS builtin takes (int4 AS1*, int4 AS3*, imm, imm). Build the
// AS-qualified pointers via integer round-trip: a generic LDS pointer's low
// 32 bits ARE the LDS byte offset per the flat-aperture rules, and generic
// global addresses equal AS1 addresses numerically.
#define AS1_V4I(p) ((__attribute__((address_space(1))) v4i*)(unsigned long long)(uintptr_t)(p))
#define AS3_V4I(p) ((__attribute__((address_space(3))) v4i*)(unsigned)(uintptr_t)(p))

// ---------------------------------------------------------------------------
// Kernel A: grid-wide scoring (memory-bound streaming of 80.6 MB).
// Stages 256-row tiles through LDS with async-to-LDS b128 copies so HBM sees
// only dense 128-bit transfers; conflict-free LDS reads (stride-25 is odd).
// ---------------------------------------------------------------------------
__global__ __launch_bounds__(256, 4) void yolo_score(
    const float* __restrict__ pred, unsigned* __restrict__ key32)
{
    __shared__ __align__(16) float s_tile[TILE_ROWS * ROW];   // 25.6 KB
    const int b = blockIdx.y, tid = threadIdx.x;
    const int r0 = blockIdx.x * TILE_ROWS;
    const int rows = min(TILE_ROWS, NPRED - r0);   // 256 or 112 (both %4==0)
    const int nf4 = (rows * ROW) >> 2;
    const float* src = pred + ((size_t)b * NPRED + r0) * ROW;

#ifdef HAVE_ASYNC_LDS
    for (int i = tid; i < nf4; i += 256) {
        __builtin_amdgcn_global_load_async_to_lds_b128(
            AS1_V4I(src + 4 * i), AS3_V4I(&s_tile[4 * i]), 0, 0);
    }
#if __has_builtin(__builtin_amdgcn_s_wait_asynccnt)
    __builtin_amdgcn_s_wait_asynccnt(0);
#else
    asm volatile("s_wait_asynccnt 0x0" ::: "memory");
#endif
#else
    for (int i = tid; i < nf4; i += 256)
        ((float4*)s_tile)[i] = ((const float4*)src)[i];
#endif
    __syncthreads();

    if (tid < rows) {
        const float* r = s_tile + tid * ROW;
        float obj = r[4];
        float mc = r[5];
        #pragma unroll
        for (int c = 1; c < NCLS; ++c) mc = fmaxf(mc, r[5 + c]);
        float conf = obj * mc;
        key32[(size_t)b * NPRED + r0 + tid] =
            (conf > CONF_T) ? __float_as_uint(conf) : 0u;   // monotonic key
    }
}

// ---------------------------------------------------------------------------
// Kernel B: per-batch top-1024 selection (16K-bin histogram + bitonic in LDS).
// Only streams 201.6 KB of keys per batch (two passes); gathers 1024 rows for emission.
// ---------------------------------------------------------------------------
__global__ __launch_bounds__(1024, 1) void yolo_select(
    const float* __restrict__ pred, const unsigned* __restrict__ key32,
    float* __restrict__ cscore, float* __restrict__ cbox,
    int* __restrict__ clabel)
{
    __shared__ unsigned s_hist[NBINS];            // 64 KB
    __shared__ unsigned s_chunk[1024];            //  4 KB
    __shared__ unsigned long long s_key[CAP];     // 16 KB
    __shared__ unsigned s_T;
    __shared__ unsigned s_cnt;

    const int b = blockIdx.x, tid = threadIdx.x;
    for (int i = tid; i < NBINS; i += 1024) s_hist[i] = 0u;
    if (tid == 0) s_cnt = 0u;
    __syncthreads();

    const unsigned* bk = key32 + (size_t)b * NPRED;
    for (int e = tid; e < NPRED; e += 1024) {
        __builtin_prefetch(bk + e + 1024, 0, 1);
        unsigned kb = bk[e];
        if (kb) atomicAdd(&s_hist[kb >> 17], 1u);
    }
    __syncthreads();

    {
        unsigned s = 0;
        #pragma unroll
        for (int q = 0; q < 16; ++q) s += s_hist[tid * 16 + q];
        s_chunk[tid] = s;
    }
    __syncthreads();
    if (tid == 0) {
        unsigned acc = 0, T = 1u;
        for (int ct = 1023; ct >= 0; --ct) {
            if (acc + s_chunk[ct] >= KSEL) {
                for (int bin = ct * 16 + 15; bin >= ct * 16; --bin) {
                    acc += s_hist[bin];
                    if (acc >= KSEL) { T = (unsigned)bin; break; }
                }
                break;
            }
            acc += s_chunk[ct];
        }
        s_T = T;
    }
    __syncthreads();
    const unsigned T = s_T;

    for (int i = tid; i < CAP; i += 1024) s_key[i] = 0ull;
    __syncthreads();
    for (int e = tid; e < NPRED; e += 1024) {
        unsigned kb = bk[e];
        if (kb && (kb >> 17) >= T) {
            unsigned pos = atomicAdd(&s_cnt, 1u);
            if (pos < CAP)
                s_key[pos] = ((unsigned long long)kb << 32) |
                             (unsigned long long)(0xFFFFFFFFu - (unsigned)e);
        }
    }
    __syncthreads();

    for (int k = 2; k <= CAP; k <<= 1) {
        for (int j = k >> 1; j > 0; j >>= 1) {
            for (int i = tid; i < CAP; i += 1024) {
                int l = i ^ j;
                if (l > i) {
                    unsigned long long a = s_key[i], c = s_key[l];
                    bool descBlk = ((i & k) == 0);
                    if (descBlk ? (a < c) : (a > c)) { s_key[i] = c; s_key[l] = a; }
                }
            }
            __syncthreads();
        }
    }

    {
        unsigned long long key = s_key[tid];
        unsigned hi = (unsigned)(key >> 32);
        unsigned eb = 0xFFFFFFFFu - (unsigned)(key & 0xFFFFFFFFull);
        float sc; int idx;
        if (hi) { sc = __uint_as_float(hi); idx = (int)eb; }
        else    { sc = 0.0f;                idx = 0; }
        const float* r = pred + ((size_t)b * NPRED + idx) * ROW;
        float x = r[0], y = r[1], w = r[2], h = r[3];
        float mc = r[5]; int lab = 0;
        #pragma unroll
        for (int c = 1; c < NCLS; ++c) {
            float v = r[5 + c];
            if (v > mc) { mc = v; lab = c; }
        }
        size_t o = (size_t)b * KSEL + tid;
        cscore[o] = sc;
        clabel[o] = lab;
        cbox[o * 4 + 0] = x - 0.5f * w;
        cbox[o * 4 + 1] = y - 0.5f * h;
        cbox[o * 4 + 2] = x + 0.5f * w;
        cbox[o * 4 + 3] = y + 0.5f * h;
    }
}

// ---------------------------------------------------------------------------
// Kernel C1: suppression bitmask, 128-row slab per block (8 x 32 = 256 blocks).
// WMMA computes area_i + area_j (rank-2 outer sum) on the XDL pipe while the
// VALU does intersection min/max. Only j > i bits are set (upper triangle).
// ---------------------------------------------------------------------------
__global__ __launch_bounds__(1024, 1) void yolo_iou(
    const float* __restrict__ cbox, const int* __restrict__ clabel,
    unsigned* __restrict__ supg)
{
    __shared__ float s_box[KSEL * 4];            // 16 KB (class-offset boxes)
    __shared__ float s_area[KSEL];               //  4 KB
    __shared__ unsigned s_slab[SLAB_ROWS * 32];  // 16 KB slab bitmask

    const int slab = blockIdx.x, b = blockIdx.y, tid = threadIdx.x;
    const int lane = tid & 31, wave = tid >> 5;
    const size_t base = (size_t)b * KSEL;
    {
        int lab = clabel[base + tid];
        float off = (float)lab * MAXWH;
        float x0 = cbox[(base + tid) * 4 + 0] + off;
        float y0 = cbox[(base + tid) * 4 + 1] + off;
        float x1 = cbox[(base + tid) * 4 + 2] + off;
        float y1 = cbox[(base + tid) * 4 + 3] + off;
        s_box[tid * 4 + 0] = x0; s_box[tid * 4 + 1] = y0;
        s_box[tid * 4 + 2] = x1; s_box[tid * 4 + 3] = y1;
        s_area[tid] = (x1 - x0) * (y1 - y0);
    }
    for (int i = tid; i < SLAB_ROWS * 32; i += 1024) s_slab[i] = 0u;
    __syncthreads();

    const int rbase = slab * SLAB_ROWS;
    for (int tt = wave; tt < 8 * 64; tt += 32) {     // 8 tile-rows x 64 tile-cols
        int ti = tt >> 6, tj = tt & 63;
        int gib = rbase + ti * 16;
        if (tj * 16 + 15 <= gib) continue;           // tile entirely below diagonal
        v2f af, bf;
        if (lane < 16) { af.x = s_area[gib + lane];      af.y = 1.0f; }
        else           { af.x = 0.0f;                    af.y = 0.0f; }
        if (lane < 16) { bf.x = 1.0f; bf.y = s_area[tj * 16 + lane]; }
        else           { bf.x = 0.0f; bf.y = 0.0f; }
        v8f acc = {};
        acc = __builtin_amdgcn_wmma_f32_16x16x4_f32(
            false, af, false, bf, (short)0, acc, false, false);

        int j = tj * 16 + (lane & 15);
        float jx0 = s_box[j * 4 + 0], jy0 = s_box[j * 4 + 1];
        float jx1 = s_box[j * 4 + 2], jy1 = s_box[j * 4 + 3];
        int mb = gib + ((lane >= 16) ? 8 : 0);
        #pragma unroll
        for (int v = 0; v < 8; ++v) {
            int i = mb + v;
            float ix0 = s_box[i * 4 + 0], iy0 = s_box[i * 4 + 1];
            float ix1 = s_box[i * 4 + 2], iy1 = s_box[i * 4 + 3];
            float lx = fmaxf(ix0, jx0), ly = fmaxf(iy0, jy0);
            float rx = fminf(ix1, jx1), ry = fminf(iy1, jy1);
            float iw = fmaxf(rx - lx, 0.0f), ih = fmaxf(ry - ly, 0.0f);
            float inter = iw * ih;
            float denom = acc[v] - inter + 1e-9f;    // (a_i + a_j) from WMMA
            if (j > i && inter > IOU_T * denom)
                atomicOr(&s_slab[(i - rbase) * 32 + (j >> 5)], 1u << (j & 31));
        }
    }
    __syncthreads();
    unsigned* dst = supg + ((size_t)b * NSLAB + slab) * (SLAB_ROWS * 32);
    for (int w = tid; w < SLAB_ROWS * 32; w += 1024) dst[w] = s_slab[w];
}

// ---------------------------------------------------------------------------
// Kernel C2: single-wave greedy NMS on the bitmask (zero barriers in the
// serial chain), clip/validate, top-300 bitonic sort, emit outputs.
// ---------------------------------------------------------------------------
__global__ __launch_bounds__(1024, 1) void yolo_finalize(
    const float* __restrict__ cscore, const float* __restrict__ cbox,
    const int* __restrict__ clabel, const unsigned* __restrict__ supg,
    const int* __restrict__ ph, const int* __restrict__ pw,
    float* __restrict__ out)
{
    __shared__ unsigned s_sup[KSEL * 32];        // 128 KB full bitmask (320KB LDS)
    __shared__ float s_bc[KSEL * 4];             // 16 KB clipped boxes
    __shared__ float s_score[KSEL];
    __shared__ int   s_label[KSEL];
    __shared__ unsigned long long s_key[KSEL];   // 8 KB
    __shared__ unsigned s_validbits[32];
    __shared__ unsigned s_keepbits[32];

    const int b = blockIdx.x, tid = threadIdx.x;
    const int lane = tid & 31, wave = tid >> 5;
    const size_t base = (size_t)b * KSEL;

    for (int w = tid; w < KSEL * 32; w += 1024)
        s_sup[w] = supg[(size_t)b * KSEL * 32 + w];

    const float W = (float)pw[0], H = (float)ph[0];
    float sc = cscore[base + tid];
    {
        float bx0 = fminf(fmaxf(cbox[(base + tid) * 4 + 0], 0.0f), W);
        float by0 = fminf(fmaxf(cbox[(base + tid) * 4 + 1], 0.0f), H);
        float bx1 = fminf(fmaxf(cbox[(base + tid) * 4 + 2], 0.0f), W);
        float by1 = fminf(fmaxf(cbox[(base + tid) * 4 + 3], 0.0f), H);
        s_bc[tid * 4 + 0] = bx0; s_bc[tid * 4 + 1] = by0;
        s_bc[tid * 4 + 2] = bx1; s_bc[tid * 4 + 3] = by1;
        s_score[tid] = sc;
        s_label[tid] = clabel[base + tid];
    }
    {   // pack candidate-validity bits, one word per wave
        unsigned long long bal = __ballot(sc > CONF_T);
        if (lane == 0) s_validbits[wave] = (unsigned)bal;
    }
    __syncthreads();

    // Serial greedy pass on a single wave: lane l holds removal bits for
    // t in [32l, 32l+32). Sup rows contain only j > i, so plain OR is exact.
    if (wave == 0) {
        unsigned remv = ~s_validbits[lane];
        for (int i = 0; i < KSEL; ++i) {
            unsigned wv = __shfl(remv, i >> 5, 32);
            if (!((wv >> (i & 31)) & 1u))
                remv |= s_sup[i * 32 + lane];
        }
        s_keepbits[lane] = ~remv;
    }
    __syncthreads();

    {
        bool keep = (s_keepbits[tid >> 5] >> (tid & 31)) & 1u;
        float bw = s_bc[tid * 4 + 2] - s_bc[tid * 4 + 0];
        float bh = s_bc[tid * 4 + 3] - s_bc[tid * 4 + 1];
        bool valid = keep && (bw > 0.0f) && (bh > 0.0f) &&
                     (bw >= 1.0f) && (bh >= 1.0f);
        s_key[tid] = valid
            ? ((((unsigned long long)__float_as_uint(s_score[tid])) << 32) |
               (unsigned long long)(unsigned)(1023 - tid))
            : (unsigned long long)(unsigned)(1023 - tid);
    }
    __syncthreads();

    for (int k = 2; k <= KSEL; k <<= 1) {
        for (int j = k >> 1; j > 0; j >>= 1) {
            int i = tid, l = i ^ j;
            if (l > i) {
                unsigned long long a = s_key[i], c = s_key[l];
                bool descBlk = ((i & k) == 0);
                if (descBlk ? (a < c) : (a > c)) { s_key[i] = c; s_key[l] = a; }
            }
            __syncthreads();
        }
    }

    if (tid < MAXDET) {
        unsigned long long key = s_key[tid];
        unsigned hi = (unsigned)(key >> 32);
        int src = 1023 - (int)(key & 1023ull);
        bool vm = (hi != 0u);
        float scf = vm ? __uint_as_float(hi) : 0.0f;
        float x0 = 0.f, y0 = 0.f, x1 = 0.f, y1 = 0.f, lab = 0.f;
        if (vm) {
            x0 = s_bc[src * 4 + 0]; y0 = s_bc[src * 4 + 1];
            x1 = s_bc[src * 4 + 2]; y1 = s_bc[src * 4 + 3];
            int li = s_label[src];
            li = li < 0 ? 0 : (li > 19 ? 19 : li);
            lab = (float)li;
        }
        float ox0 = fminf(x0, x1), ox1 = fmaxf(x0, x1);
        float oy0 = fminf(y0, y1), oy1 = fmaxf(y0, y1);
        float* boxes  = out;
        float* scores = out + (size_t)BATCH * MAXDET * 4;
        float* labels = scores + (size_t)BATCH * MAXDET;
        float* vmo    = labels + (size_t)BATCH * MAXDET;
        size_t o = (size_t)b * MAXDET + tid;
        boxes[o * 4 + 0] = ox0; boxes[o * 4 + 1] = oy0;
        boxes[o * 4 + 2] = ox1; boxes[o * 4 + 3] = oy1;
        scores[o] = scf;
        labels[o] = lab;
        vmo[o] = vm ? 1.0f : 0.0f;
    }
}

extern "C" void kernel_launch(void* const* d_in, const int* in_sizes, int n_in,
                              void* d_out, int out_size, void* d_ws, size_t ws_size,
                              hipStream_t stream) {
    (void)in_sizes; (void)n_in; (void)out_size; (void)ws_size;
    const float* pred = (const float*)d_in[0];
    const int* ph = (const int*)d_in[1];
    const int* pw = (const int*)d_in[2];
    float* out = (float*)d_out;

    char* ws = (char*)d_ws;
    unsigned* key32 = (unsigned*)ws;
    size_t o = (size_t)BATCH * NPRED * sizeof(unsigned);
    float* cscore = (float*)(ws + o); o += (size_t)BATCH * KSEL * sizeof(float);
    float* cbox   = (float*)(ws + o); o += (size_t)BATCH * KSEL * 4 * sizeof(float);
    int*   clabel = (int*)(ws + o);   o += (size_t)BATCH * KSEL * sizeof(int);
    unsigned* supg = (unsigned*)(ws + o);   // 32 * 32768 * 4 B = 4 MB

    dim3 gA((NPRED + TILE_ROWS - 1) / TILE_ROWS, BATCH);
    yolo_score<<<gA, 256, 0, stream>>>(pred, key32);
    yolo_select<<<BATCH, 1024, 0, stream>>>(pred, key32, cscore, cbox, clabel);
    dim3 gC(NSLAB, BATCH);
    yolo_iou<<<gC, 1024, 0, stream>>>(cbox, clabel, supg);
    yolo_finalize<<<BATCH, 1024, 0, stream>>>(cscore, cbox, clabel, supg,
                                              ph, pw, out);
}